// DWTLayer_84009560310481
// MI455X (gfx1250) — compile-verified
//
#include <hip/hip_runtime.h>
#include <hip/hip_bf16.h>
#include <stdint.h>

// -------- Problem constants (from reference) --------
#define NB   64          // batch
#define H    1024
#define W    1024
#define HH_  512         // H/2
#define WH_  512         // W/2
#define PLANE ((size_t)HH_ * WH_)          // 262144 elements per subband plane
#define IMG_IN ((size_t)H * W)             // 1048576 elements per input image
#define IMG_OUT (4 * PLANE)                // per-batch output elements

typedef __attribute__((ext_vector_type(4))) float v4f;

struct Quad { float ll, lh, hl, hh; };

// Haar butterfly for one 2x2 quad: a=x[2i,2j] b=x[2i,2j+1] c=x[2i+1,2j] d=x[2i+1,2j+1]
__device__ __forceinline__ Quad haar4(float a, float b, float c, float d) {
    const float s = a + b;   // row0 sum
    const float t = b - a;   // row0 diff
    const float u = c + d;   // row1 sum
    const float v = d - c;   // row1 diff
    Quad q;
    q.ll = (s + u) * 0.5f;
    q.lh = (u - s) * 0.5f;
    q.hl = (t + v) * 0.5f;
    q.hh = (v - t) * 0.5f;
    return q;
}

// =====================================================================
// Primary kernel: async-to-LDS staged Haar DWT (CDNA5 ASYNCcnt path).
// Block: 256 threads (8 waves). Each block stages 4 input rows (16 KB)
// of one batch image into LDS via 4x global_load_async_to_lds_b128 in
// GVS mode (uniform SGPR base + one 32-bit per-lane offset VGPR; the
// IOFFSET immediate advances BOTH the global and the LDS address, so a
// single vdst/vaddr pair serves all four transfers). Then it computes
// 2 output rows across all 4 subbands with coalesced non-temporal
// b128 stores.
// =====================================================================
__global__ __launch_bounds__(256) void dwt_haar_async(const float* __restrict__ in,
                                                      float* __restrict__ out) {
    __shared__ float lds[4 * W];   // 4 rows x 1024 floats = 16 KB

    const int t   = threadIdx.x;
    const int blk = blockIdx.x;          // 0 .. 64*256-1
    const int b   = blk >> 8;            // batch index
    const int oy0 = (blk & 255) << 1;    // first of 2 output rows
    const int r0  = oy0 << 1;            // first of 4 input rows

    const float* src = in + (size_t)b * IMG_IN + (size_t)r0 * W;  // uniform -> SGPR pair

    // ---- Stage 1: async copy 1024 float4 (16 KB) into LDS -------------
    // Lane i copies float4 #i of each 4 KB quarter; IOFFSET selects the
    // quarter on both the LDS side and the global side simultaneously.
    {
        const unsigned off16 = (unsigned)t << 4;                       // lane byte offset
        const unsigned loff  = (unsigned)(size_t)&lds[0] + off16;      // LDS byte address
        const unsigned goff  = off16;                                  // global byte offset vs SADDR
        asm volatile(
            "global_load_async_to_lds_b128 %0, %1, %2 offset:0 th:TH_LOAD_NT\n\t"
            "global_load_async_to_lds_b128 %0, %1, %2 offset:4096 th:TH_LOAD_NT\n\t"
            "global_load_async_to_lds_b128 %0, %1, %2 offset:8192 th:TH_LOAD_NT\n\t"
            "global_load_async_to_lds_b128 %0, %1, %2 offset:12288 th:TH_LOAD_NT\n\t"
            "s_wait_asynccnt 0x0"
            :
            : "v"(loff), "v"(goff), "s"(src)
            : "memory");
    }
    __syncthreads();   // all 8 waves' transfers visible to every lane

    // ---- Stage 2: butterfly + coalesced NT stores ---------------------
    const int oy_l = t >> 7;     // 0..1 : which of the 2 output rows
    const int xq   = t & 127;    // 4 output columns -> 8 input columns

    const v4f* row0 = (const v4f*)&lds[(oy_l * 2)     * W + xq * 8];
    const v4f* row1 = (const v4f*)&lds[(oy_l * 2 + 1) * W + xq * 8];
    const v4f r0a = row0[0], r0b = row0[1];
    const v4f r1a = row1[0], r1b = row1[1];

    const Quad q0 = haar4(r0a.x, r0a.y, r1a.x, r1a.y);
    const Quad q1 = haar4(r0a.z, r0a.w, r1a.z, r1a.w);
    const Quad q2 = haar4(r0b.x, r0b.y, r1b.x, r1b.y);
    const Quad q3 = haar4(r0b.z, r0b.w, r1b.z, r1b.w);

    const v4f LL  = { q0.ll, q1.ll, q2.ll, q3.ll };
    const v4f LH  = { q0.lh, q1.lh, q2.lh, q3.lh };
    const v4f HL  = { q0.hl, q1.hl, q2.hl, q3.hl };
    const v4f HHv = { q0.hh, q1.hh, q2.hh, q3.hh };

    float* o = out + (size_t)b * IMG_OUT + (size_t)(oy0 + oy_l) * WH_ + xq * 4;
    // Write-once streaming output: non-temporal to avoid polluting L2.
    __builtin_nontemporal_store(LL,  (v4f*)(o));
    __builtin_nontemporal_store(LH,  (v4f*)(o + PLANE));
    __builtin_nontemporal_store(HL,  (v4f*)(o + 2 * PLANE));
    __builtin_nontemporal_store(HHv, (v4f*)(o + 3 * PLANE));
}

// =====================================================================
// Reference/backup kernel: direct vectorized path (no LDS round-trip).
// Kept in the TU for codegen comparison; not launched.
// =====================================================================
__global__ __launch_bounds__(256) void dwt_haar_direct(const float* __restrict__ in,
                                                       float* __restrict__ out) {
    const size_t idx = (size_t)blockIdx.x * 256 + threadIdx.x;  // 0 .. 4194303
    const int xq = (int)(idx & 127);
    const int oy = (int)((idx >> 7) & 511);
    const int b  = (int)(idx >> 16);

    const float* src = in + (size_t)b * IMG_IN + (size_t)(2 * oy) * W + xq * 8;
    const v4f r0a = __builtin_nontemporal_load((const v4f*)src);
    const v4f r0b = __builtin_nontemporal_load((const v4f*)src + 1);
    const v4f r1a = __builtin_nontemporal_load((const v4f*)(src + W));
    const v4f r1b = __builtin_nontemporal_load((const v4f*)(src + W) + 1);

    const Quad q0 = haar4(r0a.x, r0a.y, r1a.x, r1a.y);
    const Quad q1 = haar4(r0a.z, r0a.w, r1a.z, r1a.w);
    const Quad q2 = haar4(r0b.x, r0b.y, r1b.x, r1b.y);
    const Quad q3 = haar4(r0b.z, r0b.w, r1b.z, r1b.w);

    const v4f LL  = { q0.ll, q1.ll, q2.ll, q3.ll };
    const v4f LH  = { q0.lh, q1.lh, q2.lh, q3.lh };
    const v4f HL  = { q0.hl, q1.hl, q2.hl, q3.hl };
    const v4f HHv = { q0.hh, q1.hh, q2.hh, q3.hh };

    float* o = out + (size_t)b * IMG_OUT + (size_t)oy * WH_ + xq * 4;
    __builtin_nontemporal_store(LL,  (v4f*)(o));
    __builtin_nontemporal_store(LH,  (v4f*)(o + PLANE));
    __builtin_nontemporal_store(HL,  (v4f*)(o + 2 * PLANE));
    __builtin_nontemporal_store(HHv, (v4f*)(o + 3 * PLANE));
}

extern "C" void kernel_launch(void* const* d_in, const int* in_sizes, int n_in,
                              void* d_out, int out_size, void* d_ws, size_t ws_size,
                              hipStream_t stream) {
    (void)in_sizes; (void)n_in; (void)out_size; (void)d_ws; (void)ws_size;
    const float* x = (const float*)d_in[0];
    float* y = (float*)d_out;

    // 64 batches x 256 row-pair tiles = 16384 blocks of 256 threads.
    dim3 grid(NB * (HH_ / 2));
    dim3 block(256);
    dwt_haar_async<<<grid, block, 0, stream>>>(x, y);
}